// RoIPointPool3d_7473243095656
// MI455X (gfx1250) — compile-verified
//
#include <hip/hip_runtime.h>
#include <stdint.h>

// Problem constants (match reference)
#define BATCH  4
#define NPTS   16384
#define NBOX   128
#define NCH    128
#define NSAMP  512
#define EXTRA_W 1.0f
#define FEAT   (3 + NCH)          // 131 output channels per sampled point

#define BLOCK  256                // 8 wave32 waves
#define WAVES  (BLOCK / 32)
#define TILE   1024               // points staged per LDS tile (12 KB)
#define CHUNKS (TILE / BLOCK)     // 4 ordered chunks per tile
#define NTILES (NPTS / TILE)      // 16

// ---- CDNA5 async global->LDS (ASYNCcnt-tracked, bypasses VGPRs) ----------
// GVS mode: 64-bit SGPR base + 32-bit per-lane byte offset; VDST holds the
// 32-bit LDS byte address (low 32 bits of the flat shared pointer).
__device__ __forceinline__ void async_ld_b128(uint32_t lds_addr,
                                              const float* sbase,
                                              uint32_t voff) {
    asm volatile("global_load_async_to_lds_b128 %0, %1, %2"
                 :
                 : "v"(lds_addr), "v"(voff), "s"(sbase)
                 : "memory");
}
#define S_WAIT_ASYNCCNT(n) asm volatile("s_wait_asynccnt " #n ::: "memory")

__global__ __launch_bounds__(BLOCK)
void roipool3d_kernel(const float* __restrict__ points,   // (B,N,3)
                      const float* __restrict__ feats,    // (B,N,C)
                      const float* __restrict__ boxes,    // (B,M,7)
                      float* __restrict__ pooled,         // (B,M,S,131)
                      int*   __restrict__ empty_out,      // (B,M)
                      int*   __restrict__ idx_out)        // (B,M,S)
{
    __shared__ __align__(16) float s_pts[2][TILE * 3];    // 24 KB double buffer
    __shared__ int s_idx[NSAMP];                          // first S indices in order
    __shared__ int s_sel[NSAMP];                          // wrapped selection
    __shared__ int s_wcnt[WAVES];

    const int wg   = blockIdx.x;
    const int b    = wg / NBOX;
    const int m    = wg - b * NBOX;
    const int tid  = threadIdx.x;
    const int lane = tid & 31;
    const int wv   = tid >> 5;

    // Box parameters (uniform -> scalar loads)
    const float* bx = boxes + (size_t)(b * NBOX + m) * 7;
    const float cx = bx[0], cy = bx[1], cz = bx[2];
    const float hx = bx[3] * 0.5f + EXTRA_W;
    const float hy = bx[4] * 0.5f + EXTRA_W;
    const float hz = bx[5] * 0.5f + EXTRA_W;
    const float hd = bx[6];
    // heading is in [-pi, pi]: hardware v_sin/v_cos range reduction is exact
    // enough here and avoids the ~500-instruction scalar Payne-Hanek path.
    const float ch = __cosf(hd), sh = __sinf(hd);

    // Per-thread staging slice: 12 dwords (48 B) = 3 x b128 per tile
    const uint32_t o0 = (uint32_t)tid * 48u;

    auto issue_tile = [&](int t, int buf) {
        const float* gbase = points + ((size_t)b * NPTS + (size_t)t * TILE) * 3;
        uint32_t lds = (uint32_t)(uintptr_t)(&s_pts[buf][0]);
        async_ld_b128(lds + o0,        gbase, o0);
        async_ld_b128(lds + o0 + 16u,  gbase, o0 + 16u);
        async_ld_b128(lds + o0 + 32u,  gbase, o0 + 32u);
    };

    // ---------------- Phase 1: ordered compaction of in-box indices --------
    int base = 0;                 // running count (uniform across block)
    issue_tile(0, 0);

    for (int t = 0; t < NTILES; ++t) {
        const int buf = t & 1;
        if (t + 1 < NTILES) {
            issue_tile(t + 1, buf ^ 1);   // overlap next tile's DMA
            S_WAIT_ASYNCCNT(3);           // own first 3 (tile t) complete
        } else {
            S_WAIT_ASYNCCNT(0);
        }
        __syncthreads();                  // tile t visible to all waves

        for (int ck = 0; ck < CHUNKS; ++ck) {
            const int local = ck * BLOCK + tid;       // ascending point order
            const float px = s_pts[buf][local * 3 + 0];
            const float py = s_pts[buf][local * 3 + 1];
            const float pz = s_pts[buf][local * 3 + 2];
            const float rx = px - cx, ry = py - cy, rz = pz - cz;
            const float lx = rx * ch + ry * sh;       // rotate by -heading
            const float ly = ry * ch - rx * sh;
            const bool  in = (fabsf(lx) < hx) & (fabsf(ly) < hy) & (fabsf(rz) < hz);

            const uint32_t mk = (uint32_t)__ballot(in);   // wave32 mask
            if (lane == 0) s_wcnt[wv] = __popc(mk);
            __syncthreads();

            int before = base, total = 0;
            #pragma unroll
            for (int i = 0; i < WAVES; ++i) {
                const int v = s_wcnt[i];
                total += v;
                if (i < wv) before += v;
            }
            const int pos = before + __popc(mk & ((1u << lane) - 1u));
            if (in && pos < NSAMP) s_idx[pos] = t * TILE + local;
            base += total;                 // uniform update
            __syncthreads();               // protect s_wcnt reuse / s_idx reads
            if (base >= NSAMP) break;      // uniform early exit
        }
        if (base >= NSAMP) break;
    }
    const int cnt = (base < NSAMP) ? base : NSAMP;

    // ---------------- Phase 2: wrapped selection + index/empty outputs -----
    for (int s = tid; s < NSAMP; s += BLOCK) {
        const int v = (cnt > 0) ? s_idx[s % cnt] : 0;
        s_sel[s] = v;
        __builtin_nontemporal_store(v, &idx_out[(size_t)(b * NBOX + m) * NSAMP + s]);
    }
    if (tid == 0) empty_out[b * NBOX + m] = (cnt == 0) ? 1 : 0;
    __syncthreads();

    // ---------------- Phase 3: feature gather (store-bandwidth bound) ------
    // Streaming 137 MB of output: non-temporal stores keep L2 free for the
    // gather reads (feats = 32 MB, points = 0.8 MB -> fully L2-resident).
    const size_t outBase = (size_t)(b * NBOX + m) * NSAMP * FEAT;
    const float* pbase = points + (size_t)b * NPTS * 3;
    const float* fbase = feats  + (size_t)b * NPTS * NCH;
    for (int e = tid; e < NSAMP * FEAT; e += BLOCK) {
        const int s = e / FEAT;            // const-divisor -> mulhi
        const int c = e - s * FEAT;
        float v = 0.0f;
        if (cnt > 0) {
            const int src = s_sel[s];
            v = (c < 3) ? pbase[(size_t)src * 3 + c]
                        : fbase[(size_t)src * NCH + (c - 3)];
        }
        __builtin_nontemporal_store(v, &pooled[outBase + e]);  // coalesced NT dwords
    }
}

extern "C" void kernel_launch(void* const* d_in, const int* in_sizes, int n_in,
                              void* d_out, int out_size, void* d_ws, size_t ws_size,
                              hipStream_t stream) {
    (void)in_sizes; (void)n_in; (void)out_size; (void)d_ws; (void)ws_size;
    const float* points = (const float*)d_in[0];
    const float* feats  = (const float*)d_in[1];
    const float* boxes  = (const float*)d_in[2];

    float* pooled    = (float*)d_out;
    int*   empty_out = (int*)(pooled + (size_t)BATCH * NBOX * NSAMP * FEAT);
    int*   idx_out   = empty_out + BATCH * NBOX;

    roipool3d_kernel<<<BATCH * NBOX, BLOCK, 0, stream>>>(
        points, feats, boxes, pooled, empty_out, idx_out);
}